// HMM_68264210202822
// MI455X (gfx1250) — compile-verified
//
#include <hip/hip_runtime.h>

// HMM scaled-forward via chunked associative scan, matmuls on the CDNA5
// f32 WMMA pipe (V_WMMA_F32_16X16X4_F32).  Y fixed at 64 (4x4 tiles of 16x16).
//
// K1: per-chunk 64x64 transfer matrices (one WMMA matmul per timestep)
// K2: group products of 64 chunk matrices (async-LDS staged when available)
// K3: serial scan over group matrices (tiny)
// K4: per-group scan -> chunk-entry alpha vectors
// K5: 16 chunks batched per block -> the serial per-step matvec becomes a
//     16x64 @ 64x64 WMMA matmul; per-row normalize; coalesced row stores.

typedef float v2f __attribute__((ext_vector_type(2)));
typedef float v8f __attribute__((ext_vector_type(8)));
typedef int hmm_v4i __attribute__((vector_size(16)));

#define YD 64
#define PITCH 66  // LDS row pitch (floats); even -> 8B-aligned paired A-frag loads

#if defined(__has_builtin)
#if __has_builtin(__builtin_amdgcn_global_load_async_to_lds_b128) && \
    __has_builtin(__builtin_amdgcn_s_wait_asynccnt)
#define HMM_HAVE_ASYNC 1
#endif
#endif
#ifndef HMM_HAVE_ASYNC
#define HMM_HAVE_ASYNC 0
#endif

// Builtin signature (from hipcc diagnostics): first arg is
// "int __vector_size(16) __device__*"  == int4 addrspace(1)*  (non-const).
#define GLB1(p) ((__attribute__((address_space(1))) hmm_v4i*)(p))
#define LDS3(p) ((__attribute__((address_space(3))) hmm_v4i*)(p))

static __device__ __forceinline__ v8f v8zero() {
  v8f z = {0.f, 0.f, 0.f, 0.f, 0.f, 0.f, 0.f, 0.f};
  return z;
}

static __device__ __forceinline__ v8f wmma_f32_k4(v2f a, v2f b, v8f c) {
  return __builtin_amdgcn_wmma_f32_16x16x4_f32(false, a, false, b, (short)0, c,
                                               false, false);
}

// DPP row-rotate (within each 16-lane row): pure VALU, no LDS round-trips.
template <int CTRL>
static __device__ __forceinline__ float dpp_mov_f(float v) {
  return __int_as_float(
      __builtin_amdgcn_update_dpp(0, __float_as_int(v), CTRL, 0xf, 0xf, true));
}
static __device__ __forceinline__ float row_sum16(float v) {
  v += dpp_mov_f<0x121>(v);  // row_ror:1
  v += dpp_mov_f<0x122>(v);  // row_ror:2
  v += dpp_mov_f<0x124>(v);  // row_ror:4
  v += dpp_mov_f<0x128>(v);  // row_ror:8
  return v;
}
static __device__ __forceinline__ float row_max16(float v) {
  v = fmaxf(v, dpp_mov_f<0x121>(v));
  v = fmaxf(v, dpp_mov_f<0x122>(v));
  v = fmaxf(v, dpp_mov_f<0x124>(v));
  v = fmaxf(v, dpp_mov_f<0x128>(v));
  return v;
}

// ---------------------------------------------------------------------------
// K1: P_c = prod_{t in chunk} (T * diag(b[:,x[t]])), with per-step max rescale.
// ---------------------------------------------------------------------------
__global__ void __launch_bounds__(512)
k1_chunk_mats(const int* __restrict__ x, const float* __restrict__ tr,
              const float* __restrict__ b, float* __restrict__ P,
              int T, int X, int L)
{
  __shared__ float sP[2][YD * PITCH];
  __shared__ __attribute__((aligned(16))) float swmax[32];

  const int tid = threadIdx.x;
  const int w = tid >> 5, l = tid & 31;
  const int ti = w >> 2, tj = w & 3;
  const int lh = l >> 4, ll = l & 15;
  const int col = 16 * tj + ll;

  v2f bfrag[16];
#pragma unroll
  for (int kk = 0; kk < 16; ++kk) {
    const int k0 = 4 * kk + 2 * lh;
    bfrag[kk].x = tr[(k0 + 0) * YD + col];
    bfrag[kk].y = tr[(k0 + 1) * YD + col];
  }

  for (int i = tid; i < YD * YD; i += 512) {
    const int r = i >> 6, cc = i & 63;
    sP[0][r * PITCH + cc] = (r == cc) ? 1.0f : 0.0f;
  }
  __syncthreads();

  const int c = blockIdx.x;
  const int t0 = c * L;
  const int nstep = min(L, (T - 1) - t0);
  int cur = 0;

  // software-pipelined emission chase: x[t] -> b[col*X + x[t]]
  float e = 0.f;
  if (nstep >= 1) e = b[(size_t)col * X + x[t0 + 1]];

  for (int s = 1; s <= nstep; ++s) {
    // issue next x load before the matmul so the L2 chase overlaps compute
    int xn = 0;
    if (s < nstep) xn = x[t0 + s + 1];
    __builtin_prefetch(x + min(t0 + s + 64, T - 1), 0, 1);

    v8f acc0 = v8zero(), acc1 = v8zero();
#pragma unroll
    for (int kk = 0; kk < 8; ++kk) {
      const int k0 = 4 * kk + 2 * lh;
      const int k1 = 4 * (kk + 8) + 2 * lh;
      v2f a0, a1;
      a0.x = sP[cur][(16 * ti + ll) * PITCH + k0];
      a0.y = sP[cur][(16 * ti + ll) * PITCH + k0 + 1];
      a1.x = sP[cur][(16 * ti + ll) * PITCH + k1];
      a1.y = sP[cur][(16 * ti + ll) * PITCH + k1 + 1];
      acc0 = wmma_f32_k4(a0, bfrag[kk], acc0);
      acc1 = wmma_f32_k4(a1, bfrag[kk + 8], acc1);
    }
    v8f acc = acc0 + acc1;

    float en = 0.f;
    if (s < nstep) en = b[(size_t)col * X + xn];

#pragma unroll
    for (int r = 0; r < 8; ++r) acc[r] *= e;

    // block max: 7 lane-local max, DPP row reduce, LDS combine of 32 halves
    float m = acc[0];
#pragma unroll
    for (int r = 1; r < 8; ++r) m = fmaxf(m, acc[r]);
    m = row_max16(m);
    if (ll == 0) swmax[2 * w + lh] = m;
    __syncthreads();
    const float4* m4 = reinterpret_cast<const float4*>(swmax);
    float bm = 0.f;
#pragma unroll
    for (int q = 0; q < 8; ++q) {
      const float4 v4 = m4[q];
      bm = fmaxf(bm, fmaxf(fmaxf(v4.x, v4.y), fmaxf(v4.z, v4.w)));
    }
    const float inv = 1.0f / bm;

    const int nxt = cur ^ 1;
#pragma unroll
    for (int r = 0; r < 8; ++r)
      sP[nxt][(16 * ti + r + 8 * lh) * PITCH + col] = acc[r] * inv;
    __syncthreads();
    cur = nxt;
    e = en;
  }

  float* Pc = P + (size_t)c * (YD * YD);
  for (int i = tid; i < YD * YD; i += 512)
    Pc[i] = sP[cur][(i >> 6) * PITCH + (i & 63)];
}

// ---------------------------------------------------------------------------
// K2: group products G_g = P_{g*GS} @ ... @ P_{g*GS+GS-1}.
// Next chunk matrix staged into LDS with async-to-LDS loads when available.
// ---------------------------------------------------------------------------
__global__ void __launch_bounds__(512)
k2_group_mats(const float* __restrict__ P, float* __restrict__ G, int GS)
{
  __shared__ float sP[2][YD * PITCH];
  __shared__ __attribute__((aligned(16))) float swmax[32];
#if HMM_HAVE_ASYNC
  __shared__ __attribute__((aligned(16))) float sS[2][YD * YD];
#endif

  const int tid = threadIdx.x;
  const int w = tid >> 5, l = tid & 31;
  const int ti = w >> 2, tj = w & 3;
  const int lh = l >> 4, ll = l & 15;
  const int col = 16 * tj + ll;
  const int g = blockIdx.x;

  const float* P0 = P + (size_t)g * GS * (YD * YD);
  for (int i = tid; i < YD * YD; i += 512)
    sP[0][(i >> 6) * PITCH + (i & 63)] = P0[i];
#if HMM_HAVE_ASYNC
  if (GS > 1) {
    const float* Pn = P0 + (YD * YD);
    __builtin_amdgcn_global_load_async_to_lds_b128(
        GLB1(Pn + tid * 4), LDS3(&sS[0][tid * 4]), 0, 0);
    __builtin_amdgcn_global_load_async_to_lds_b128(
        GLB1(Pn + 2048 + tid * 4), LDS3(&sS[0][2048 + tid * 4]), 0, 0);
    __builtin_amdgcn_s_wait_asynccnt(0);
  }
#endif
  __syncthreads();

  int cur = 0;
  int scur = 0;
  for (int step = 1; step < GS; ++step) {
    const int snxt = scur ^ 1;
#if HMM_HAVE_ASYNC
    if (step + 1 < GS) {  // prefetch next B matrix while we compute
      const float* Pn2 = P0 + (size_t)(step + 1) * (YD * YD);
      __builtin_amdgcn_global_load_async_to_lds_b128(
          GLB1(Pn2 + tid * 4), LDS3(&sS[snxt][tid * 4]), 0, 0);
      __builtin_amdgcn_global_load_async_to_lds_b128(
          GLB1(Pn2 + 2048 + tid * 4), LDS3(&sS[snxt][2048 + tid * 4]), 0, 0);
    }
#else
    const float* Pn = P0 + (size_t)step * (YD * YD);
#endif
    v8f acc0 = v8zero(), acc1 = v8zero();
#pragma unroll
    for (int kk = 0; kk < 8; ++kk) {
      const int k0 = 4 * kk + 2 * lh;
      const int k1 = 4 * (kk + 8) + 2 * lh;
      v2f a0, a1, b0, b1;
      a0.x = sP[cur][(16 * ti + ll) * PITCH + k0];
      a0.y = sP[cur][(16 * ti + ll) * PITCH + k0 + 1];
      a1.x = sP[cur][(16 * ti + ll) * PITCH + k1];
      a1.y = sP[cur][(16 * ti + ll) * PITCH + k1 + 1];
#if HMM_HAVE_ASYNC
      b0.x = sS[scur][(k0 + 0) * YD + col];
      b0.y = sS[scur][(k0 + 1) * YD + col];
      b1.x = sS[scur][(k1 + 0) * YD + col];
      b1.y = sS[scur][(k1 + 1) * YD + col];
#else
      b0.x = Pn[(k0 + 0) * YD + col];
      b0.y = Pn[(k0 + 1) * YD + col];
      b1.x = Pn[(k1 + 0) * YD + col];
      b1.y = Pn[(k1 + 1) * YD + col];
#endif
      acc0 = wmma_f32_k4(a0, b0, acc0);
      acc1 = wmma_f32_k4(a1, b1, acc1);
    }
    v8f acc = acc0 + acc1;

    float m = acc[0];
#pragma unroll
    for (int r = 1; r < 8; ++r) m = fmaxf(m, acc[r]);
    m = row_max16(m);
    if (ll == 0) swmax[2 * w + lh] = m;
    __syncthreads();
    const float4* m4 = reinterpret_cast<const float4*>(swmax);
    float bm = 0.f;
#pragma unroll
    for (int q = 0; q < 8; ++q) {
      const float4 v4 = m4[q];
      bm = fmaxf(bm, fmaxf(fmaxf(v4.x, v4.y), fmaxf(v4.z, v4.w)));
    }
    const float inv = 1.0f / bm;

    const int nxt = cur ^ 1;
#pragma unroll
    for (int r = 0; r < 8; ++r)
      sP[nxt][(16 * ti + r + 8 * lh) * PITCH + col] = acc[r] * inv;
#if HMM_HAVE_ASYNC
    __builtin_amdgcn_s_wait_asynccnt(0);  // own staged loads done (overlapped)
#endif
    __syncthreads();
    cur = nxt;
    scur = snxt;
  }

  float* Gg = G + (size_t)g * (YD * YD);
  for (int i = tid; i < YD * YD; i += 512)
    Gg[i] = sP[cur][(i >> 6) * PITCH + (i & 63)];
}

// ---------------------------------------------------------------------------
// K3: serial scan over NG group matrices (single block).  Emits alpha row 0.
// ---------------------------------------------------------------------------
__global__ void __launch_bounds__(256)
k3_group_scan(const int* __restrict__ x, const float* __restrict__ b,
              const float* __restrict__ pi, const float* __restrict__ G,
              float* __restrict__ gstart, float* __restrict__ out,
              int X, int NG)
{
  __shared__ float v[64];
  __shared__ float part[4][64];
  __shared__ float sG[64 * 64];
  const int tid = threadIdx.x;
  const int j = tid & 63, kg = tid >> 6;

  if (tid < 64) v[tid] = b[(size_t)tid * X + x[0]] * pi[tid];
  __syncthreads();
  float tot = 0.f;
  for (int k = 0; k < 64; ++k) tot += v[k];
  const float inv0 = 1.0f / tot;
  __syncthreads();
  if (tid < 64) {
    const float a0 = v[tid] * inv0;
    v[tid] = a0;
    out[tid] = a0;
    gstart[tid] = a0;
  }
  __syncthreads();

  for (int g = 0; g < NG; ++g) {
    const float* Gg = G + (size_t)g * 4096;
    for (int i = tid; i < 4096; i += 256) sG[i] = Gg[i];
    __syncthreads();
    float s = 0.f;
    for (int k = 0; k < 16; ++k) {
      const int kk = kg * 16 + k;
      s += v[kk] * sG[kk * 64 + j];
    }
    part[kg][j] = s;
    __syncthreads();
    const float nv = part[0][j] + part[1][j] + part[2][j] + part[3][j];
    __syncthreads();
    if (kg == 0) part[0][j] = nv;
    __syncthreads();
    float t2 = 0.f;
    for (int k = 0; k < 64; ++k) t2 += part[0][k];
    const float invn = 1.0f / t2;
    __syncthreads();
    if (kg == 0) {
      const float nn = nv * invn;
      v[j] = nn;
      gstart[(size_t)(g + 1) * 64 + j] = nn;
    }
    __syncthreads();
  }
}

// ---------------------------------------------------------------------------
// K4: within each group, scan chunk matrices -> chunk-entry alphas cstart[c].
// ---------------------------------------------------------------------------
__global__ void __launch_bounds__(256)
k4_chunk_scan(const float* __restrict__ P, const float* __restrict__ gstart,
              float* __restrict__ cstart, int GS)
{
  __shared__ float v[64];
  __shared__ float part[4][64];
  __shared__ float sP[64 * 64];
  const int tid = threadIdx.x;
  const int j = tid & 63, kg = tid >> 6;
  const int g = blockIdx.x;

  if (tid < 64) v[tid] = gstart[(size_t)g * 64 + tid];
  __syncthreads();

  for (int i2 = 0; i2 < GS; ++i2) {
    const int c = g * GS + i2;
    if (kg == 0) cstart[(size_t)c * 64 + j] = v[j];
    const float* Pc = P + (size_t)c * 4096;
    for (int i = tid; i < 4096; i += 256) sP[i] = Pc[i];
    __syncthreads();
    float s = 0.f;
    for (int k = 0; k < 16; ++k) {
      const int kk = kg * 16 + k;
      s += v[kk] * sP[kk * 64 + j];
    }
    part[kg][j] = s;
    __syncthreads();
    const float nv = part[0][j] + part[1][j] + part[2][j] + part[3][j];
    __syncthreads();
    if (kg == 0) part[0][j] = nv;
    __syncthreads();
    float t2 = 0.f;
    for (int k = 0; k < 64; ++k) t2 += part[0][k];
    const float invn = 1.0f / t2;
    __syncthreads();
    if (kg == 0) v[j] = nv * invn;
    __syncthreads();
  }
}

// ---------------------------------------------------------------------------
// K5: batched alpha fill; 16 chunks per block advance together as a 16x64
// WMMA matmul per timestep.  4 waves; wave w owns column tile w.
// ---------------------------------------------------------------------------
__global__ void __launch_bounds__(128)
k5_alpha_fill(const int* __restrict__ x, const float* __restrict__ tr,
              const float* __restrict__ b, const float* __restrict__ cstart,
              float* __restrict__ out, int T, int X, int L)
{
  __shared__ float sA[2][16 * PITCH];
  __shared__ __attribute__((aligned(16))) float spart[64];  // [M][w]

  const int tid = threadIdx.x;
  const int w = tid >> 5, l = tid & 31;
  const int lh = l >> 4, ll = l & 15;
  const int j = 16 * w + ll;
  const int c0 = blockIdx.x * 16;

  v2f bfrag[16];
#pragma unroll
  for (int kk = 0; kk < 16; ++kk) {
    const int k0 = 4 * kk + 2 * lh;
    bfrag[kk].x = tr[(k0 + 0) * YD + j];
    bfrag[kk].y = tr[(k0 + 1) * YD + j];
  }

  for (int i = tid; i < 16 * 64; i += 128) {
    const int r = i >> 6, cc = i & 63;
    sA[0][r * PITCH + cc] = cstart[(size_t)(c0 + r) * 64 + cc];
  }
  __syncthreads();

  // software-pipelined emission values for this lane's 8 batch rows
  float ecur[8];
#pragma unroll
  for (int r = 0; r < 8; ++r) {
    const int t = (c0 + r + 8 * lh) * L + 1;
    ecur[r] = b[(size_t)j * X + x[min(t, T - 1)]];
  }

  int cur = 0;
  for (int s = 1; s <= L; ++s) {
    // start next-step x chase before the matmul
    int xn[8];
#pragma unroll
    for (int r = 0; r < 8; ++r) xn[r] = 0;
    if (s < L) {
#pragma unroll
      for (int r = 0; r < 8; ++r) {
        const int t = (c0 + r + 8 * lh) * L + s + 1;
        xn[r] = x[min(t, T - 1)];
      }
    }

    v8f acc0 = v8zero(), acc1 = v8zero();
#pragma unroll
    for (int kk = 0; kk < 8; ++kk) {
      const int k0 = 4 * kk + 2 * lh;
      const int k1 = 4 * (kk + 8) + 2 * lh;
      v2f a0, a1;
      a0.x = sA[cur][ll * PITCH + k0];
      a0.y = sA[cur][ll * PITCH + k0 + 1];
      a1.x = sA[cur][ll * PITCH + k1];
      a1.y = sA[cur][ll * PITCH + k1 + 1];
      acc0 = wmma_f32_k4(a0, bfrag[kk], acc0);
      acc1 = wmma_f32_k4(a1, bfrag[kk + 8], acc1);
    }
    v8f acc = acc0 + acc1;

    float enx[8];
#pragma unroll
    for (int r = 0; r < 8; ++r) enx[r] = 0.f;
    if (s < L) {
#pragma unroll
      for (int r = 0; r < 8; ++r) enx[r] = b[(size_t)j * X + xn[r]];
    }

#pragma unroll
    for (int r = 0; r < 8; ++r) acc[r] *= ecur[r];

    // per-row sums: DPP rotate-reduce within each 16-lane half, then combine
    // the 4 column tiles through a tiny LDS array read back as float4.
#pragma unroll
    for (int r = 0; r < 8; ++r) {
      const float t2 = row_sum16(acc[r]);
      if (ll == 0) spart[(r + 8 * lh) * 4 + w] = t2;
    }
    __syncthreads();
    const int nxt = cur ^ 1;
#pragma unroll
    for (int r = 0; r < 8; ++r) {
      const int M = r + 8 * lh;
      const float4 p4 = *reinterpret_cast<const float4*>(&spart[M * 4]);
      const float tot = (p4.x + p4.y) + (p4.z + p4.w);
      const float a = acc[r] / tot;
      const int t = (c0 + M) * L + s;
      if (t <= T - 1) out[(size_t)t * 64 + j] = a;  // coalesced 64B segments
      sA[nxt][M * PITCH + j] = a;
    }
    __syncthreads();
    cur = nxt;
#pragma unroll
    for (int r = 0; r < 8; ++r) ecur[r] = enx[r];
  }
}

// ---------------------------------------------------------------------------
extern "C" void kernel_launch(void* const* d_in, const int* in_sizes, int n_in,
                              void* d_out, int out_size, void* d_ws, size_t ws_size,
                              hipStream_t stream) {
  (void)n_in; (void)out_size;
  const int* x = (const int*)d_in[0];
  const float* tr = (const float*)d_in[1];
  const float* b = (const float*)d_in[2];
  const float* pi = (const float*)d_in[3];
  float* out = (float*)d_out;

  const int T = in_sizes[0];
  const int X = in_sizes[2] / YD;

  const size_t per_chunk = 16384 + 256 + (16384 + 256) / 64 + 16;
  long nc = (long)(ws_size / per_chunk);
  if (nc > 8192) nc = 8192;
  nc -= nc % 64;
  if (nc < 64) nc = 64;
  const int NC = (int)nc;
  const int GS = 64;
  const int NG = NC / GS;
  int L = (T - 1 + NC - 1) / NC;
  if (L < 1) L = 1;

  float* P = (float*)d_ws;                          // NC * 4096
  float* G = P + (size_t)NC * 4096;                 // NG * 4096
  float* gstart = G + (size_t)NG * 4096;            // (NG+1) * 64
  float* cstart = gstart + (size_t)(NG + 1) * 64;   // NC * 64

  k1_chunk_mats<<<NC, 512, 0, stream>>>(x, tr, b, P, T, X, L);
  k2_group_mats<<<NG, 512, 0, stream>>>(P, G, GS);
  k3_group_scan<<<1, 256, 0, stream>>>(x, b, pi, G, gstart, out, X, NG);
  k4_chunk_scan<<<NG, 256, 0, stream>>>(P, gstart, cstart, GS);
  k5_alpha_fill<<<NC / 16, 128, 0, stream>>>(x, tr, b, cstart, out, T, X, L);
}